// AffineToDenseShift_60705067762008
// MI455X (gfx1250) — compile-verified
//
#include <hip/hip_runtime.h>

// CDNA5 / gfx1250. Wave32. Uses V_WMMA_F32_16X16X4_F32: D(16x16) = A(16x4) x B(4x16) + C.
// One WMMA computes 64 output voxels (3 channels each) by replicating the 3x4
// affine rows 4x across the 16 A-rows with w-offsets folded into the
// translation column. Kernel is store-bandwidth bound (330 MB out, ~14us floor
// at 23.3 TB/s), so stores are coalesced non-temporal dwords.

typedef __attribute__((ext_vector_type(2))) float v2f;
typedef __attribute__((ext_vector_type(8))) float v8f;

#define BATCH 4
#define DD 160
#define HH 192
#define WW 224
#define WAVES_PER_BLOCK 8
#define THREADS_PER_BLOCK (WAVES_PER_BLOCK * 32)
#define NROWS (BATCH * DD * HH)  // 122880, divisible by WAVES_PER_BLOCK

__global__ __launch_bounds__(THREADS_PER_BLOCK)
void AffineToDenseShift_wmma_kernel(const float* __restrict__ mat,
                                    float* __restrict__ out)
{
    const int lane = threadIdx.x & 31;
    const int waveInBlock = threadIdx.x >> 5;
    const int row = blockIdx.x * WAVES_PER_BLOCK + waveInBlock; // flat (b,d,h)

    const int b = row / (DD * HH);
    const int rem = row - b * (DD * HH);
    const int d = rem / HH;
    const int h = rem - d * HH;

    // ---------- A operand: 16x4 f32 matrix, 2 VGPRs ----------
    // ISA layout: lanes 0-15 hold row M=lane, cols {0,1} in {v0,v1};
    //             lanes 16-31 hold row M=lane-16, cols {2,3} in {v0,v1}.
    // Row assignment (channel c = row%3 within each triple):
    //   rows 0-2  : w-offset  0     rows 8-10 : w-offset 16
    //   rows 3-5  : w-offset 32     rows 11-13: w-offset 48
    //   rows 6,7,14,15 : zero (unused)
    // Cols 0..2 = E[c][:] = A[c][:] - I[c][:]  (out = loc - mesh)
    // Col  3    = t[c] + offs * E[c][2]        (w-offset folded via homogeneous 1)
    const int M  = lane & 15;
    const int m7 = M & 7;
    const bool rowValid = (m7 < 6);
    const int c = m7 % 3;
    const float offs = (float)((M >= 8 ? 16 : 0) + (m7 >= 3 ? 32 : 0));

    const float* mrow = mat + b * 12 + c * 4;   // [B,3,4] row for channel c
    float e0 = mrow[0] - (c == 0 ? 1.0f : 0.0f);
    float e1 = mrow[1] - (c == 1 ? 1.0f : 0.0f);
    float e2 = mrow[2] - (c == 2 ? 1.0f : 0.0f);
    float tc = mrow[3] + offs * e2;

    const bool lo = (lane < 16);
    float a0 = lo ? e0 : e2;   // col 0 (low) / col 2 (high)
    float a1 = lo ? e1 : tc;   // col 1 (low) / col 3 (high)
    if (!rowValid) { a0 = 0.0f; a1 = 0.0f; }
    v2f Aop; Aop.x = a0; Aop.y = a1;

    const float md = (float)d - (float)(DD - 1) * 0.5f;
    const float mh = (float)h - (float)(HH - 1) * 0.5f;
    const float cw = (float)(WW - 1) * 0.5f;

    float* rowout = out + (size_t)row * (WW * 3);

    for (int wb = 0; wb < WW; wb += 64) {
        // ---------- B operand: 4x16 f32, 2 VGPRs ----------
        // VGPR0: lanes 0-15 = row K=0 (md), lanes 16-31 = row K=2 (mw0+N)
        // VGPR1: lanes 0-15 = row K=1 (mh), lanes 16-31 = row K=3 (1.0)
        const float mw0 = (float)wb - cw;
        v2f Bop;
        Bop.x = lo ? md : (mw0 + (float)(lane - 16));
        Bop.y = lo ? mh : 1.0f;

        v8f acc = {0.f, 0.f, 0.f, 0.f, 0.f, 0.f, 0.f, 0.f};
        // D[M][N] = E[c] . (md, mh, mw0+N) + t[c] + offs*E[c][2]
        acc = __builtin_amdgcn_wmma_f32_16x16x4_f32(
            /*neg_a=*/false, Aop, /*neg_b=*/false, Bop,
            /*c_mod=*/(short)0, acc, /*reuse_a=*/false, /*reuse_b=*/false);

        // Lane l holds channels 0..2 of voxel wb+l in acc[0..2]
        // and of voxel wb+32+l in acc[3..5] -> contiguous 12B per group,
        // consecutive lanes -> consecutive voxels (dense 384B span per wave).
        float* p0 = rowout + (size_t)(wb + lane) * 3;
        __builtin_nontemporal_store(acc[0], p0 + 0);
        __builtin_nontemporal_store(acc[1], p0 + 1);
        __builtin_nontemporal_store(acc[2], p0 + 2);

        if (wb + 32 < WW) {  // wave-uniform; skipped only for wb == 192
            float* p1 = rowout + (size_t)(wb + 32 + lane) * 3;
            __builtin_nontemporal_store(acc[3], p1 + 0);
            __builtin_nontemporal_store(acc[4], p1 + 1);
            __builtin_nontemporal_store(acc[5], p1 + 2);
        }
    }
}

extern "C" void kernel_launch(void* const* d_in, const int* in_sizes, int n_in,
                              void* d_out, int out_size, void* d_ws, size_t ws_size,
                              hipStream_t stream) {
    (void)in_sizes; (void)n_in; (void)out_size; (void)d_ws; (void)ws_size;
    const float* mat = (const float*)d_in[0];   // [4,3,4] float32
    float* out = (float*)d_out;                 // [4,160,192,224,3] float32

    const int blocks = NROWS / WAVES_PER_BLOCK; // 15360, exact -> EXEC all-1s
    AffineToDenseShift_wmma_kernel<<<blocks, THREADS_PER_BLOCK, 0, stream>>>(mat, out);
}